// Proposed_model_64596308132176
// MI455X (gfx1250) — compile-verified
//
#include <hip/hip_runtime.h>
#include <hip/hip_bf16.h>

#define DD 128  // feature dim

typedef float v2f __attribute__((ext_vector_type(2)));
typedef float v8f __attribute__((ext_vector_type(8)));

struct Scales8 { float s[8]; };

// ---------------------------------------------------------------------------
// elementwise helpers
// ---------------------------------------------------------------------------
__global__ void zero_kernel(float* __restrict__ p, long n) {
    long i = blockIdx.x * (long)blockDim.x + threadIdx.x;
    if (i < n) p[i] = 0.0f;
}

__global__ void deg_count_kernel(const int* __restrict__ idx, int E,
                                 float* __restrict__ deg) {
    long i = blockIdx.x * (long)blockDim.x + threadIdx.x;
    if (i < E) atomicAdd(deg + idx[i], 1.0f);
}

__global__ void inv_sqrt_kernel(float* __restrict__ d, int n) {
    long i = blockIdx.x * (long)blockDim.x + threadIdx.x;
    if (i < n) d[i] = rsqrtf(fmaxf(d[i], 1.0f));
}

// y[row][c] = x[row][c] * s1[row] * (s2 ? s2[row] : 1)   (float4 granularity)
__global__ void scale_rows_kernel(const float* __restrict__ x,
                                  const float* __restrict__ s1,
                                  const float* __restrict__ s2,
                                  float* __restrict__ y, int nrows) {
    long i = blockIdx.x * (long)blockDim.x + threadIdx.x;
    long total = (long)nrows * (DD / 4);
    if (i >= total) return;
    long row = i / (DD / 4);
    float sc = s1[row];
    if (s2) sc *= s2[row];
    float4 v = ((const float4*)x)[i];
    v.x *= sc; v.y *= sc; v.z *= sc; v.w *= sc;
    ((float4*)y)[i] = v;
}

// per edge: agg[dst[e]] += xs[src[e]] * (ew ? ew[e] : 1); one wave-lane per 4 ch
__global__ void edge_scatter_kernel(const float* __restrict__ xs,
                                    const int* __restrict__ src,
                                    const int* __restrict__ dst,
                                    const float* __restrict__ ew,
                                    float* __restrict__ agg, int E) {
    long gid = blockIdx.x * (long)blockDim.x + threadIdx.x;
    int e = (int)(gid >> 5);
    if (e >= E) return;
    int c = ((int)gid & 31) * 4;
    float4 v = *(const float4*)(xs + (long)src[e] * DD + c);
    float w = ew ? ew[e] : 1.0f;
    float* p = agg + (long)dst[e] * DD + c;
    atomicAdd(p + 0, v.x * w);
    atomicAdd(p + 1, v.y * w);
    atomicAdd(p + 2, v.z * w);
    atomicAdd(p + 3, v.w * w);
}

// out = wa*ha + wo*ho + wself*hg   (float4 granularity)
__global__ void combine_kernel(const float* __restrict__ ha,
                               const float* __restrict__ ho,
                               const float* __restrict__ hg,
                               float* __restrict__ out, long n4,
                               float wa, float wo, float wself) {
    long i = blockIdx.x * (long)blockDim.x + threadIdx.x;
    if (i >= n4) return;
    float4 a = ((const float4*)ha)[i];
    float4 o = ((const float4*)ho)[i];
    float4 g = ((const float4*)hg)[i];
    float4 r;
    r.x = wa * a.x + wo * o.x + wself * g.x;
    r.y = wa * a.y + wo * o.y + wself * g.y;
    r.z = wa * a.z + wo * o.z + wself * g.z;
    r.w = wa * a.w + wo * o.w + wself * g.w;
    ((float4*)out)[i] = r;
}

// ---------------------------------------------------------------------------
// Fused layer attention:
//   scores[l][n][e] = (t_scaled[l][n][:] @ W)[e] * a[e],   t_scaled = t * ps[l]
//   w = softmax over l;  out[n][e] = sum_l t_scaled[l][n][e] * w[l][n][e]
// One wave per 16x16 output tile; L accumulator tile-sets held in registers.
// Uses V_WMMA_F32_16X16X4_F32 (fp32 A/B/C, K-loop of 32 steps).
// nnodes must be a multiple of 16 (30000 is); EXEC stays all-1s (no tails).
// ---------------------------------------------------------------------------
template <int L>
__global__ __launch_bounds__(256) void attn_kernel(
    const float* __restrict__ t,     // [L][nnodes][DD]
    const float* __restrict__ W,     // [DD][DD]
    const float* __restrict__ avec,  // [DD]
    float* __restrict__ out,         // [nnodes][DD]
    int nnodes, Scales8 ps) {
    const int lane = threadIdx.x & 31;
    const int wave = threadIdx.x >> 5;
    const int tile = blockIdx.x * 8 + wave;     // 8 waves / block
    const int ncolTiles = DD / 16;              // 8
    const int m16 = (tile / ncolTiles) * 16;    // node-tile base
    const int n16 = (tile % ncolTiles) * 16;    // col-tile base
    const int m = lane & 15;
    const int hi = lane >> 4;

    const v8f zero8 = {0.f, 0.f, 0.f, 0.f, 0.f, 0.f, 0.f, 0.f};
    v8f acc[L];
#pragma unroll
    for (int l = 0; l < L; ++l) acc[l] = zero8;

    // A-fragment row pointer per layer: this lane supplies matrix row m16+m
    const float* trow[L];
#pragma unroll
    for (int l = 0; l < L; ++l)
        trow[l] = t + ((long)l * nnodes + m16 + m) * DD;

    for (int k0 = 0; k0 < DD; k0 += 4) {
        const int kk = k0 + 2 * hi;  // lanes 0-15: K={k0,k0+1}; 16-31: K={k0+2,k0+3}
        v2f B;
        B.x = W[(long)(kk + 0) * DD + n16 + m];
        B.y = W[(long)(kk + 1) * DD + n16 + m];
#pragma unroll
        for (int l = 0; l < L; ++l) {
            v2f A;
            A.x = trow[l][kk + 0];
            A.y = trow[l][kk + 1];
            acc[l] = __builtin_amdgcn_wmma_f32_16x16x4_f32(
                false, A, false, B, (short)0, acc[l], false, false);
        }
    }

    // C/D layout: VGPR r -> (row = m16 + r + 8*hi, col = n16 + (lane&15))
    const int col = n16 + m;
    const float av = avec[col];
#pragma unroll
    for (int r = 0; r < 8; ++r) {
        const int row = m16 + r + 8 * hi;
        float s[L];
        float mx = -3.4e38f;
#pragma unroll
        for (int l = 0; l < L; ++l) {
            s[l] = acc[l][r] * ps.s[l] * av;
            mx = fmaxf(mx, s[l]);
        }
        float sum = 0.f;
#pragma unroll
        for (int l = 0; l < L; ++l) {
            s[l] = __expf(s[l] - mx);
            sum += s[l];
        }
        const float inv = 1.0f / sum;
        float o = 0.f;
#pragma unroll
        for (int l = 0; l < L; ++l) {
            const float tv = t[((long)l * nnodes + row) * DD + col] * ps.s[l];
            o += tv * (s[l] * inv);
        }
        out[(long)row * DD + col] = o;
    }
}

// ---------------------------------------------------------------------------
// host-side orchestration
// ---------------------------------------------------------------------------
extern "C" void kernel_launch(void* const* d_in, const int* in_sizes, int n_in,
                              void* d_out, int out_size, void* d_ws, size_t ws_size,
                              hipStream_t stream) {
    const int NU = 50000, NI = 30000;
    const int E_UG = in_sizes[0];
    const int E_AND = in_sizes[2];
    const int E_OR = in_sizes[8];

    const int* ug_u = (const int*)d_in[0];
    const int* ug_g = (const int*)d_in[1];
    const int* and_src[3] = {(const int*)d_in[2], (const int*)d_in[4], (const int*)d_in[6]};
    const int* and_dst[3] = {(const int*)d_in[3], (const int*)d_in[5], (const int*)d_in[7]};
    const int* or_src = (const int*)d_in[8];
    const int* or_dst = (const int*)d_in[9];
    const float* user_emb = (const float*)d_in[10];
    const float* item_emb = (const float*)d_in[11];
    const float* W_and = (const float*)d_in[12];
    const float* a_and = (const float*)d_in[13];
    const float* W_or = (const float*)d_in[14];
    const float* a_or = (const float*)d_in[15];
    const float* weight_edge = (const float*)d_in[16];
    const float* weight_node = (const float*)d_in[17];

    // ---- workspace carve-out (floats, 256B aligned slices) ----
    float* ws = (float*)d_ws;
    size_t off = 0;
    auto alloc = [&](size_t n) {
        float* p = ws + off;
        off += (n + 63) & ~(size_t)63;
        return p;
    };
    float* inv_u = alloc(NU);
    float* inv_g = alloc(NI);
    float* io[3]; float* ii[3];
    for (int i = 0; i < 3; ++i) { io[i] = alloc(NI); ii[i] = alloc(NI); }
    float* or_out_n = alloc(NI);
    float* or_in_n = alloc(NI);
    float* xs = alloc((size_t)NU * DD);        // scaled-source scratch (max rows)
    float* huA = alloc((size_t)NU * DD);       // hu after iter 1
    float* hgA = alloc((size_t)NI * DD);       // hg after iter 1; later h_and
    float* hgB = alloc((size_t)NI * DD);       // hg after iter 2
    const size_t SZ = (size_t)NI * DD;
    float* and_stack = alloc(7 * SZ);          // [7][NI][DD]
    float* or_stack = alloc(4 * SZ);           // [4][NI][DD]

    float* hu_out = (float*)d_out;                      // [NU][DD]
    float* hgame_out = (float*)d_out + (size_t)NU * DD; // [NI][DD]

    const int TB = 256;
    auto blocks1 = [&](long n) { return (unsigned)((n + TB - 1) / TB); };

    // ---- degree norms ----
    auto inv_sqrt_deg = [&](const int* idx, int E, float* outp, int n) {
        zero_kernel<<<blocks1(n), TB, 0, stream>>>(outp, n);
        deg_count_kernel<<<blocks1(E), TB, 0, stream>>>(idx, E, outp);
        inv_sqrt_kernel<<<blocks1(n), TB, 0, stream>>>(outp, n);
    };
    inv_sqrt_deg(ug_u, E_UG, inv_u, NU);
    inv_sqrt_deg(ug_g, E_UG, inv_g, NI);
    for (int i = 0; i < 3; ++i) {
        inv_sqrt_deg(and_src[i], E_AND, io[i], NI);
        inv_sqrt_deg(and_dst[i], E_AND, ii[i], NI);
    }
    inv_sqrt_deg(or_src, E_OR, or_out_n, NI);
    inv_sqrt_deg(or_dst, E_OR, or_in_n, NI);

    // ---- gcn: out = inv_in * scatter_dst( (x*inv_out*extra)[src] * ew ) ----
    auto gcn = [&](const float* x, const float* inv_outv, const float* extra,
                   const int* src, const int* dst, const float* ew, int E,
                   int n_src, int n_dst, const float* inv_inv, float* outp) {
        scale_rows_kernel<<<blocks1((long)n_src * (DD / 4)), TB, 0, stream>>>(
            x, inv_outv, extra, xs, n_src);
        zero_kernel<<<blocks1((long)n_dst * DD), TB, 0, stream>>>(outp, (long)n_dst * DD);
        edge_scatter_kernel<<<blocks1((long)E * 32), TB, 0, stream>>>(
            xs, src, dst, ew, outp, E);
        scale_rows_kernel<<<blocks1((long)n_dst * (DD / 4)), TB, 0, stream>>>(
            outp, inv_inv, nullptr, outp, n_dst);
    };

    // ---- user-game propagation, 2 iterations ----
    // iter 1: from (user_emb, item_emb)
    gcn(item_emb, inv_g, weight_node, ug_g, ug_u, weight_edge, E_UG, NI, NU, inv_u, huA);
    gcn(user_emb, inv_u, nullptr, ug_u, ug_g, nullptr, E_UG, NU, NI, inv_g, hgA);
    // iter 2: from (huA, hgA); hu2 goes straight to d_out
    gcn(hgA, inv_g, weight_node, ug_g, ug_u, weight_edge, E_UG, NI, NU, inv_u, hu_out);
    gcn(huA, inv_u, nullptr, ug_u, ug_g, nullptr, E_UG, NU, NI, inv_g, hgB);

    // ---- AND stacks: slot0=item_emb, slots 1-3 = layer1, slots 4-6 = layer2 ----
    hipMemcpyAsync(and_stack, item_emb, SZ * sizeof(float), hipMemcpyDeviceToDevice, stream);
    for (int i = 0; i < 3; ++i)
        gcn(and_stack, io[i], nullptr, and_src[i], and_dst[i], nullptr, E_AND,
            NI, NI, ii[i], and_stack + (1 + i) * SZ);
    for (int i = 0; i < 3; ++i)
        gcn(and_stack + (1 + i) * SZ, io[i], nullptr, and_src[i], and_dst[i], nullptr,
            E_AND, NI, NI, ii[i], and_stack + (4 + i) * SZ);

    Scales8 ps_and = {{1.f, 1.f, 1.f, 1.f, 1.f, 1.f, 1.f, 1.f}};
    const unsigned attn_blocks = (unsigned)((NI / 16) * (DD / 16) / 8);  // 1875
    float* h_and = hgA;  // reuse (dead after iter 2)
    attn_kernel<7><<<attn_blocks, 256, 0, stream>>>(and_stack, W_and, a_and, h_and, NI, ps_and);

    // ---- OR stack: slot0=item_emb, slots 1-3 = h1,h2,h3 (unscaled; decay in attn) ----
    hipMemcpyAsync(or_stack, item_emb, SZ * sizeof(float), hipMemcpyDeviceToDevice, stream);
    for (int li = 1; li <= 3; ++li)
        gcn(or_stack + (li - 1) * SZ, or_out_n, nullptr, or_src, or_dst, nullptr,
            E_OR, NI, NI, or_in_n, or_stack + li * SZ);

    // p_li = max(1 - (3-li)*0.2, 0.2) -> {0.6, 0.8, 1.0}; slot0 scale = 1
    Scales8 ps_or = {{1.f, 0.6f, 0.8f, 1.f, 1.f, 1.f, 1.f, 1.f}};
    float* h_or = huA;  // reuse (dead after hg2)
    attn_kernel<4><<<attn_blocks, 256, 0, stream>>>(or_stack, W_or, a_or, h_or, NI, ps_or);

    // ---- final combine: h_game = w_and*h_and + w_or*h_or + w_self*hg2 ----
    const float GAMMA = 80.0f;
    const float w_or_c = GAMMA / (GAMMA + 2.0f);
    const float w_and_c = w_or_c / GAMMA;
    const float w_self_c = w_or_c / GAMMA;
    combine_kernel<<<blocks1(SZ / 4), TB, 0, stream>>>(
        h_and, h_or, hgB, hgame_out, (long)(SZ / 4), w_and_c, w_or_c, w_self_c);
}